// GraphSAGE_36661840839215
// MI455X (gfx1250) — compile-verified
//
#include <hip/hip_runtime.h>

typedef float v2f __attribute__((ext_vector_type(2)));
typedef float v8f __attribute__((ext_vector_type(8)));

// ---------------------------------------------------------------------------
// zero a float buffer (grid-stride)
// ---------------------------------------------------------------------------
__global__ void zero_kernel(float* __restrict__ p, long long n) {
    long long i = (long long)blockIdx.x * blockDim.x + threadIdx.x;
    long long stride = (long long)gridDim.x * blockDim.x;
    for (; i < n; i += stride) p[i] = 0.0f;
}

// ---------------------------------------------------------------------------
// degree: one thread per edge, deg[dst] += 1
// ---------------------------------------------------------------------------
__global__ void degree_kernel(const long long* __restrict__ dst,
                              float* __restrict__ deg, int n_edges) {
    int e = blockIdx.x * blockDim.x + threadIdx.x;
    if (e < n_edges) atomicAdd(&deg[(int)dst[e]], 1.0f);
}

// ---------------------------------------------------------------------------
// scatter-add: agg[dst] += feat[src], D=128. 32 lanes per edge, each lane
// moves one float4 (coalesced 128-bit global load, 4 f32 atomics).
// ---------------------------------------------------------------------------
__global__ void scatter_kernel(const long long* __restrict__ src,
                               const long long* __restrict__ dst,
                               const float* __restrict__ feat,  // [N,128]
                               float* __restrict__ agg,         // [N,128]
                               int n_edges) {
    int tid = blockIdx.x * blockDim.x + threadIdx.x;
    int e = tid >> 5;
    if (e >= n_edges) return;
    int c = tid & 31;                      // 32 lanes * 4 floats = 128 dims
    long long s = src[e], d = dst[e];
    const float4 v = *(const float4*)(feat + s * 128 + c * 4);
    float* o = agg + d * 128 + c * 4;
    atomicAdd(o + 0, v.x);
    atomicAdd(o + 1, v.y);
    atomicAdd(o + 2, v.z);
    atomicAdd(o + 3, v.w);
}

// ---------------------------------------------------------------------------
// Fused SAGE linear: out = (agg * 1/max(deg,1)) @ wl^T + xin @ wr^T + bias
// One wave (32 threads) computes one 16x16 output tile with
// V_WMMA_F32_16X16X4_F32, stepping K by 4.  N_nodes % 16 == 0 so EXEC is
// all-ones everywhere (WMMA requirement).
//
// A 16x4 f32 layout:  M = lane&15, VGPR j holds K = k0 + 2*(lane>>4) + j
// B 4x16  f32 layout:  N = lane&15, VGPR j holds K = k0 + 2*(lane>>4) + j
// C/D 16x16 layout:    VGPR r: M = r + 8*(lane>>4), N = lane&15
// ---------------------------------------------------------------------------
template <int K_DIM, int N_OUT, bool RELU>
__global__ __launch_bounds__(32)
void sage_gemm_kernel(const float* __restrict__ agg,   // [N,K_DIM] (summed)
                      const float* __restrict__ deg,   // [N]
                      const float* __restrict__ xin,   // [N,K_DIM]
                      const float* __restrict__ wl,    // [N_OUT,K_DIM]
                      const float* __restrict__ wr,    // [N_OUT,K_DIM]
                      const float* __restrict__ bias,  // [N_OUT]
                      float* __restrict__ out)         // [N,N_OUT]
{
    const int lane = threadIdx.x & 31;
    const int lo = lane & 15;
    const int hi = lane >> 4;                 // 0 or 1
    const long long row = (long long)blockIdx.x * 16 + lo;   // node for A
    const int col = blockIdx.y * 16 + lo;                    // out-dim for B

    const float inv = 1.0f / fmaxf(deg[row], 1.0f);

    const float* agg_p = agg + row * K_DIM;
    const float* x_p   = xin + row * K_DIM;
    const float* wl_p  = wl + (long long)col * K_DIM;
    const float* wr_p  = wr + (long long)col * K_DIM;

    v8f acc = {};
#pragma unroll
    for (int k0 = 0; k0 < K_DIM; k0 += 4) {
        const int ka = k0 + 2 * hi;
        v2f a_agg, a_x, b_l, b_r;
        a_agg.x = agg_p[ka + 0] * inv;
        a_agg.y = agg_p[ka + 1] * inv;
        a_x.x   = x_p[ka + 0];
        a_x.y   = x_p[ka + 1];
        b_l.x   = wl_p[ka + 0];
        b_l.y   = wl_p[ka + 1];
        b_r.x   = wr_p[ka + 0];
        b_r.y   = wr_p[ka + 1];
        acc = __builtin_amdgcn_wmma_f32_16x16x4_f32(
            false, a_agg, false, b_l, (short)0, acc, false, false);
        acc = __builtin_amdgcn_wmma_f32_16x16x4_f32(
            false, a_x, false, b_r, (short)0, acc, false, false);
    }

    const float bv = bias[col];
    const long long mbase = (long long)blockIdx.x * 16 + hi * 8;
#pragma unroll
    for (int r = 0; r < 8; ++r) {
        float v = acc[r] + bv;
        if (RELU) v = fmaxf(v, 0.0f);
        out[(mbase + r) * N_OUT + col] = v;
    }
}

// ---------------------------------------------------------------------------
// launcher
// ---------------------------------------------------------------------------
extern "C" void kernel_launch(void* const* d_in, const int* in_sizes, int n_in,
                              void* d_out, int out_size, void* d_ws, size_t ws_size,
                              hipStream_t stream) {
    constexpr int IN_DIM = 128, HID_DIM = 128, OUT_DIM = 64;

    const float*     x    = (const float*)d_in[0];
    const long long* eidx = (const long long*)d_in[1];   // int64 [2,E]
    const float*     wl1  = (const float*)d_in[2];
    const float*     wr1  = (const float*)d_in[3];
    const float*     b1   = (const float*)d_in[4];
    const float*     wl2  = (const float*)d_in[5];
    const float*     wr2  = (const float*)d_in[6];
    const float*     b2   = (const float*)d_in[7];
    float* out = (float*)d_out;

    const int N = in_sizes[0] / IN_DIM;      // 100000 (divisible by 16)
    const int E = in_sizes[1] / 2;           // 1600000
    const long long* src = eidx;
    const long long* dst = eidx + E;

    // workspace: deg[N] | agg[N*128] | h[N*128]
    float* deg = (float*)d_ws;
    float* agg = deg + N;
    float* h   = agg + (long long)N * HID_DIM;

    const long long aggN = (long long)N * HID_DIM;

    // ---- layer 1 ----
    zero_kernel<<<2048, 256, 0, stream>>>(deg, (long long)N);
    zero_kernel<<<4096, 256, 0, stream>>>(agg, aggN);
    degree_kernel<<<(E + 255) / 256, 256, 0, stream>>>(dst, deg, E);
    {
        long long threads = (long long)E * 32;
        scatter_kernel<<<(int)((threads + 255) / 256), 256, 0, stream>>>(
            src, dst, x, agg, E);
    }
    {
        dim3 grid(N / 16, HID_DIM / 16);
        sage_gemm_kernel<IN_DIM, HID_DIM, true>
            <<<grid, 32, 0, stream>>>(agg, deg, x, wl1, wr1, b1, h);
    }

    // ---- layer 2 ----
    zero_kernel<<<4096, 256, 0, stream>>>(agg, aggN);
    {
        long long threads = (long long)E * 32;
        scatter_kernel<<<(int)((threads + 255) / 256), 256, 0, stream>>>(
            src, dst, h, agg, E);
    }
    {
        dim3 grid(N / 16, OUT_DIM / 16);
        sage_gemm_kernel<HID_DIM, OUT_DIM, false>
            <<<grid, 32, 0, stream>>>(agg, deg, h, wl2, wr2, b2, out);
    }
}